// GaussianBlur_6631429505654
// MI455X (gfx1250) — compile-verified
//
#include <hip/hip_runtime.h>
#include <hip/hip_bf16.h>

// Depthwise 65x65 Gaussian blur, separable, streamed down full-height column
// strips. Both 1D passes are GEMMs against a banded weight matrix executed
// with V_WMMA_F32_16X16X4_F32 (fp32 in / fp32 accumulate, matching reference).
//
// Per block: 64-wide x 512-tall output strip, 32-row chunks through a 128-row
// circular LDS mid buffer. Every input row is read once (2x x-halo only):
// ~192MB HBM traffic (~8.6us at 23.3 TB/s). The banded weight matrix is
// materialized once as a dense 16x80 LDS panel so the WMMA inner loops are
// pure ds_load_b64 + v_wmma (no per-iteration clamp/select chains), and
// global loads for chunk c+2 are issued into registers before vert(c) to
// hide HBM latency behind the matrix pipe.

#define KS       65
#define KHALF    32
#define IMG      512
#define SW       64                 // strip width (output cols per block)
#define CH       32                 // rows per chunk
#define LCOLS    (SW + 2*KHALF)     // 128 haloed input cols
#define LSTR     132                // 128 + pad: 132%64=4 -> 16 rows on distinct banks
#define MROWS    128                // circular mid buffer rows (4 chunks, power of 2)
#define MSTR     68                 // 64 + pad
#define BTSTR    84                 // 80 + pad: 84%64=20 -> 16 rows on distinct banks
#define KB_COUNT 20                 // ceil(80/4) K-blocks of K=4
#define NCH      (IMG / CH)         // 16 output chunks per strip

typedef __attribute__((ext_vector_type(2))) float v2f;
typedef __attribute__((ext_vector_type(4))) float v4f;
typedef __attribute__((ext_vector_type(8))) float v8f;

// banded weight lookup: w[i] if 0 <= i <= 64 else 0 (used only to build bt panel)
__device__ __forceinline__ float wband(const float* __restrict__ wn, int i) {
    int ic = i < 0 ? 0 : (i > (KS - 1) ? (KS - 1) : i);
    float v = wn[ic];
    return (i >= 0 && i <= (KS - 1)) ? v : 0.0f;
}

struct ChunkRegs { v4f q[4]; };   // 32x128 chunk: 16 floats per thread, in flight

__global__ __launch_bounds__(256) void gauss_blur_wmma_stream(const float* __restrict__ x,
                                                              const float* __restrict__ sigma_p,
                                                              float* __restrict__ out) {
    __shared__ __align__(16) float lin[CH * LSTR];      // one 32x128 input chunk
    __shared__ __align__(16) float midb[MROWS * MSTR];  // circular horizontal-pass buffer
    __shared__ __align__(16) float bt[16 * BTSTR];      // dense banded panel: bt[n][j] = wn[j-n]
    __shared__ float wraw[KS];
    __shared__ float wn[KS];

    const int tid  = threadIdx.x;
    const int lane = tid & 31;
    const int wave = tid >> 5;
    const int n16  = lane & 15;   // M (rows, A) or N (cols, B/C/D) within 16-wide tile
    const int half = lane >> 4;   // selects K pair / upper-M half per WMMA layout

    const int ox0 = blockIdx.x * SW;
    const int gx0 = ox0 - KHALF;
    const bool edge = !(gx0 >= 0 && gx0 + LCOLS <= IMG);   // block-uniform
    const size_t ibase = (size_t)blockIdx.y * (size_t)(IMG * IMG);
    const float* img_in  = x + ibase;
    float*       img_out = out + ibase;

    // per-wave tile assignment (uniform within a wave -> EXEC all-1s at WMMAs)
    const int wrb = wave >> 2;    // row-block (0..1) within a 32-row chunk
    const int wxb = wave & 3;     // col-block (0..3) within the 64-wide strip

    // --- issue global loads of chunk m into registers (no LDS touch) ---
    auto issue = [&](int m) -> ChunkRegs {
        ChunkRegs cr;
        const int gy0 = m * CH;
        if (!edge) {
            #pragma unroll
            for (int k = 0; k < 4; ++k) {
                int idx = tid + k * 256;
                int r = idx >> 5, c4 = (idx & 31) << 2;   // 32 float4 per row
                int gy = gy0 + r;
                v4f v = {0.f, 0.f, 0.f, 0.f};
                if (gy < IMG) v = *(const v4f*)&img_in[gy * IMG + gx0 + c4];
                cr.q[k] = v;
            }
        } else {
            #pragma unroll
            for (int k = 0; k < 4; ++k) {
                int idx = tid + k * 256;
                int r = idx >> 5, c4 = (idx & 31) << 2;
                int gy = gy0 + r;
                v4f v = {0.f, 0.f, 0.f, 0.f};
                #pragma unroll
                for (int e = 0; e < 4; ++e) {
                    int gx = gx0 + c4 + e;
                    if (gy < IMG && (unsigned)gx < (unsigned)IMG)
                        v[e] = img_in[gy * IMG + gx];
                }
                cr.q[k] = v;
            }
        }
        return cr;
    };

    // --- drain the registers into the lin chunk buffer ---
    auto store_lin = [&](const ChunkRegs& cr) {
        #pragma unroll
        for (int k = 0; k < 4; ++k) {
            int idx = tid + k * 256;
            int r = idx >> 5, c4 = (idx & 31) << 2;
            *(v4f*)&lin[r * LSTR + c4] = cr.q[k];
        }
    };

    // --- horizontal pass for chunk m: mid_g[32m+r][x] = sum_k wn[k]*lin[r][x+k]
    //     C = A(16 rows of lin x 80) * B(80 x 16 banded), 1 tile per wave ---
    auto horiz = [&](int m) {
        v8f c = {0.f, 0.f, 0.f, 0.f, 0.f, 0.f, 0.f, 0.f};
        const int abase = (wrb * 16 + n16) * LSTR + wxb * 16;
        const int bbase = n16 * BTSTR;
        #pragma unroll
        for (int kb = 0; kb < KB_COUNT; ++kb) {
            int j0 = kb * 4 + half * 2;
            v2f a = *(const v2f*)&lin[abase + j0];     // A[m][j0], A[m][j0+1]
            v2f b = *(const v2f*)&bt[bbase + j0];      // B[j][n] = wn[j-n]
            c = __builtin_amdgcn_wmma_f32_16x16x4_f32(false, a, false, b,
                                                      (short)0, c, false, false);
        }
        int brow0 = (m & 3) * CH + wrb * 16 + half * 8;   // circular slot, no wrap within chunk
        int bcol  = wxb * 16 + n16;
        #pragma unroll
        for (int r = 0; r < 8; ++r)
            midb[(brow0 + r) * MSTR + bcol] = c[r];
    };

    // --- vertical pass for output chunk cc: out[y][x] = sum_k wn[k]*mid_g[y+k-32][x]
    //     C = A(16 x 80 banded) * B(80 x 16 of circular mid), 1 tile per wave ---
    auto vert = [&](int cc) {
        const int ybase = cc * CH + wrb * 16;
        const int col   = wxb * 16 + n16;
        const int abase = n16 * BTSTR;
        v8f c = {0.f, 0.f, 0.f, 0.f, 0.f, 0.f, 0.f, 0.f};
        #pragma unroll
        for (int kb = 0; kb < KB_COUNT; ++kb) {
            int j0 = kb * 4 + half * 2;
            v2f a = *(const v2f*)&bt[abase + j0];               // A[m][j] = wn[j-m]
            int g0 = (ybase - KHALF + j0)     & (MROWS - 1);    // circular rows
            int g1 = (ybase - KHALF + j0 + 1) & (MROWS - 1);
            v2f b = { midb[g0 * MSTR + col], midb[g1 * MSTR + col] };
            c = __builtin_amdgcn_wmma_f32_16x16x4_f32(false, a, false, b,
                                                      (short)0, c, false, false);
        }
        int gy = ybase + half * 8;
        int gx = ox0 + col;
        #pragma unroll
        for (int r = 0; r < 8; ++r)
            img_out[(size_t)(gy + r) * IMG + gx] = c[r];
    };

    // --- prologue: weights, zero pad slot, banded panel, prime the pipeline ---
    ChunkRegs cr = issue(0);                            // chunk 0 loads in flight
    if (tid < KS) {
        float sig = sigma_p[0];
        float t = -1.0f + (float)tid * (2.0f / (float)(KS - 1));
        wraw[tid] = expf(-(t * t) / (2.0f * sig * sig));
    }
    for (int i = tid; i < CH * SW; i += 256) {          // zero slot: rows 96..127 (g in [-32,0))
        int r = i >> 6, c0 = i & 63;
        midb[(96 + r) * MSTR + c0] = 0.0f;
    }
    __syncthreads();
    if (tid < KS) {                                     // normalize (same sum everywhere)
        float s = 0.0f;
        #pragma unroll
        for (int i = 0; i < KS; ++i) s += wraw[i];
        wn[tid] = wraw[tid] / s;
    }
    __syncthreads();
    for (int i = tid; i < 16 * 80; i += 256) {          // dense banded panel
        int n = i / 80, j = i - n * 80;
        bt[n * BTSTR + j] = wband(wn, j - n);
    }
    store_lin(cr);                                      // lin := chunk 0
    cr = issue(1);
    __syncthreads();                                    // bt + lin(0) ready
    horiz(0);
    __syncthreads();                                    // mid slot 0 ready; lin consumed

    // --- steady-state pipeline (2 barriers per chunk) ---
    for (int c = 0; c < NCH; ++c) {
        store_lin(cr);                                  // lin := chunk c+1 (chunk 16 = zeros)
        if (c + 2 <= NCH) cr = issue(c + 2);            // overlap HBM latency with vert(c)
        __syncthreads();                                // lin ready; orders vert(c-1) vs horiz(c+1)
        horiz(c + 1);                                   // writes slot (c+1)&3 (outside vert window)
        __syncthreads();                                // mid window [c-1, c+1] complete
        vert(c);                                        // reads slots (c-1..c+1)&3
    }
}

extern "C" void kernel_launch(void* const* d_in, const int* in_sizes, int n_in,
                              void* d_out, int out_size, void* d_ws, size_t ws_size,
                              hipStream_t stream) {
    (void)n_in; (void)out_size; (void)d_ws; (void)ws_size;
    const float* x     = (const float*)d_in[0];
    const float* sigma = (const float*)d_in[1];
    float* out = (float*)d_out;

    int nimg = in_sizes[0] / (IMG * IMG);   // N*C = 64 depthwise images
    dim3 grid(IMG / SW, nimg, 1);           // 8 strips x 64 images = 512 blocks
    gauss_blur_wmma_stream<<<grid, 256, 0, stream>>>(x, sigma, out);
}